// Stationary_Capsules_1d_23991687316117
// MI455X (gfx1250) — compile-verified
//
#include <hip/hip_runtime.h>

// Stationary capsules: s = (z + beta) * rsqrt(conv3x3x1_wrap(u^2) + eps)
// B=4, C = 8(t) * 16(c) * 16(d) = 2048, H*W = 4096.
//
// Strategy (MI455X, gfx1250):
//  - Pure bandwidth-bound (>=402 MB traffic, ~0.7 GFLOP): read u once via the
//    Tensor Data Mover into LDS (3D strided tile, LDS pad for bank-conflict-free
//    reads), read z once, write s once.  23.3 TB/s floor ~= 17 us.
//  - The 9-tap wrap stencil over (t,d) is expressed as banded 16x16 circulant
//    matrices applied along d via V_WMMA_F32_16X16X4_F32, with the t-wrap
//    handled by routing each source row t' into 3 accumulator tiles.

#define NT        8
#define NCAPS     16
#define CAPD      16
#define CSTRIDE   (NCAPS * CAPD)        // channel stride per t step = 256
#define HW        4096
#define BATCH     4
#define PTILE     128                   // p elements per workgroup
#define PITCH     136                   // 128 data + 8 pad dwords per (t,d) row
#define EPS       1e-6f

typedef float v2f  __attribute__((ext_vector_type(2)));
typedef float v8f  __attribute__((ext_vector_type(8)));
typedef unsigned int u32x4 __attribute__((ext_vector_type(4)));
typedef int   i32x8 __attribute__((ext_vector_type(8)));
typedef int   i32x4 __attribute__((ext_vector_type(4)));

__global__ __launch_bounds__(256) void caps_stencil_kernel(
    const float* __restrict__ z, const float* __restrict__ u,
    const float* __restrict__ wt, const float* __restrict__ beta_p,
    float* __restrict__ out)
{
  __shared__ float sh[NT * CAPD * PITCH];   // 8*16*136*4 = 69632 B

  const int pt = blockIdx.x;          // p tile index (0..31)
  const int c  = blockIdx.y;          // capsule (0..15)
  const int b  = blockIdx.z;          // batch (0..3)
  const int p0 = pt * PTILE;

  const int tid  = threadIdx.x;
  const int lane = tid & 31;
  const int wv   = tid >> 5;          // wave 0..7 -> 16-wide p slice
  const int lh   = lane >> 4;         // lane half (0/1)
  const int ln   = lane & 15;

  const size_t chan_base = (size_t)(b * (NT * CSTRIDE) + c * CAPD);
  // element (t, d, x) of the tile lives at usrc[(t*256 + d)*HW + x]
  const float* usrc = u + chan_base * HW + (size_t)p0;

  // ---------------- stage u tile (8t x 16d x 128p) into LDS ----------------
#if __has_builtin(__builtin_amdgcn_tensor_load_to_lds)
  if (wv == 0) {
    unsigned long long ga = (unsigned long long)(uintptr_t)usrc;
    unsigned int lds_off  = (unsigned int)(uintptr_t)&sh[0]; // flat[31:0] == LDS byte offset

    // D# group 0 (CDNA5 ISA)
    u32x4 g0;
    g0[0] = 1u;                                            // count=1 (valid), user mode
    g0[1] = lds_off;                                       // lds_addr
    g0[2] = (unsigned int)(ga & 0xFFFFFFFFu);              // global_addr[31:0]
    g0[3] = (unsigned int)((ga >> 32) & 0x01FFFFFFu)       // global_addr[56:32]
          | (2u << 30);                                    // type = 2 ("image")

    // D# group 1 (section 8.4)
    i32x8 g1;
    g1[0] = (2 << 16)                  // data_size = 4 bytes
          | (1 << 20)                  // pad_enable (LDS pitch padding)
          | (6 << 22)                  // pad_interval: every 128 dwords
          | (7 << 25);                 // pad_amount: 8 dwords -> pitch 136
    g1[1] = 0;                         // atomic_barrier_addr=0; tensor_dim0[15:0]=0
    g1[2] = 0x4000;                    // tensor_dim0[31:16] (dim0 = 2^30, never OOB); tensor_dim1[15:0]=0
    g1[3] = 0x4000 | (PTILE << 16);    // tensor_dim1[31:16] (2^30); tile_dim0 = 128
    g1[4] = CAPD | (NT << 16);         // tile_dim1 = 16, tile_dim2 = 8
    g1[5] = HW;                        // tensor_dim0_stride = 4096 elements (d step)
    g1[6] = 0;                         // stride0[47:32]=0; stride1[15:0]=0
    g1[7] = (CSTRIDE * HW) >> 16;      // tensor_dim1_stride = 1048576 (t step) -> [47:16]

    // D# groups 2/3 (section 8.5/8.6): 3D tile, tile_dim3 = 0
    i32x4 g2; g2[0] = 0x40000000; g2[1] = 0; g2[2] = 0; g2[3] = 0; // tensor_dim2 large
    i32x4 g3; g3[0] = 0; g3[1] = 0; g3[2] = 0; g3[3] = 0;
    i32x8 gx; gx[0]=0; gx[1]=0; gx[2]=0; gx[3]=0; gx[4]=0; gx[5]=0; gx[6]=0; gx[7]=0;

    __builtin_amdgcn_tensor_load_to_lds(g0, g1, g2, g3, gx, 0);
    __builtin_amdgcn_s_wait_tensorcnt(0);
  }
#else
  // Fallback: cooperative vectorized staging through VGPRs.
  for (int i = tid; i < (NT * CAPD * PTILE) / 4; i += 256) {
    int row = i >> 5;                    // (t*16 + d), 32 float4 per row
    int xv  = i & 31;
    int t = row >> 4, d = row & 15;
    const float* gp = usrc + (size_t)(t * CSTRIDE + d) * HW + xv * 4;
    float4 val = *(const float4*)gp;
    *(float4*)&sh[row * PITCH + xv * 4] = val;
  }
#endif
  __syncthreads();

  // ---------------- build stencil A-matrix fragments ----------------
  // A_a[d,d'] = w[a,b] where b = (d' - d + 1) mod 16 if b in {0,1,2}, else 0.
  float w9[9];
#pragma unroll
  for (int i = 0; i < 9; ++i) w9[i] = wt[i];
  const float beta = beta_p[0];

  // A frag layout (16x4 f32): lane -> M = ln; VGPR j -> K = 4k + 2*lh + j.
  v2f afr[3][4];
#pragma unroll
  for (int a = 0; a < 3; ++a)
#pragma unroll
    for (int k = 0; k < 4; ++k)
#pragma unroll
      for (int j = 0; j < 2; ++j) {
        int dp   = 4 * k + 2 * lh + j;         // d' column within full 16
        int bsel = (dp - ln + 1) & 15;
        afr[a][k][j] = (bsel < 3) ? w9[a * 3 + bsel] : 0.0f;
      }

  v8f acc[NT];
  const v8f zero8 = {0.f, 0.f, 0.f, 0.f, 0.f, 0.f, 0.f, 0.f};
#pragma unroll
  for (int t = 0; t < NT; ++t) acc[t] = zero8;

  const int pl = wv * 16 + ln;   // this lane's p column inside the 128-p tile

  // ---------------- WMMA stencil: acc[(t'-a+1)&7] += A_a * Usq[t'] ----------------
#pragma unroll
  for (int tp = 0; tp < NT; ++tp) {
    // B frag (4x16 f32): lane -> N = ln; VGPR j -> K = 2*lh + j (within chunk k)
    v2f bfr[4];
#pragma unroll
    for (int k = 0; k < 4; ++k)
#pragma unroll
      for (int j = 0; j < 2; ++j) {
        int dp = 4 * k + 2 * lh + j;
        float uv = sh[(tp * CAPD + dp) * PITCH + pl];
        bfr[k][j] = uv * uv;                   // square on LDS read
      }
#pragma unroll
    for (int a = 0; a < 3; ++a) {
      int tt = (tp - a + 1) & 7;               // target output row t
#pragma unroll
      for (int k = 0; k < 4; ++k) {
        acc[tt] = __builtin_amdgcn_wmma_f32_16x16x4_f32(
            false, afr[a][k], false, bfr[k], (short)0, acc[tt], false, false);
      }
    }
  }

  // ---------------- epilogue: s = (z + beta) * rsqrt(v + eps) ----------------
  const size_t hw_off = (size_t)p0 + pl;
#pragma unroll
  for (int t = 0; t < NT; ++t) {
#pragma unroll
    for (int r = 0; r < 8; ++r) {
      int d = r + 8 * lh;                      // D-tile row mapping
      size_t gidx = (chan_base + (size_t)t * CSTRIDE + d) * HW + hw_off;
      float v = acc[t][r];
      out[gidx] = (z[gidx] + beta) * rsqrtf(v + EPS);
    }
  }
}

extern "C" void kernel_launch(void* const* d_in, const int* in_sizes, int n_in,
                              void* d_out, int out_size, void* d_ws, size_t ws_size,
                              hipStream_t stream) {
  const float* z    = (const float*)d_in[0];
  const float* u    = (const float*)d_in[1];
  const float* wt   = (const float*)d_in[2];   // (1,1,3,3,1) -> 9 floats
  const float* beta = (const float*)d_in[3];   // 1 float
  float* out = (float*)d_out;

  dim3 grid(HW / PTILE, NCAPS, BATCH);         // 32 x 16 x 4 = 2048 blocks
  caps_stencil_kernel<<<grid, 256, 0, stream>>>(z, u, wt, beta, out);
}